// RGCNN_model_4294967296037
// MI455X (gfx1250) — compile-verified
//
#include <hip/hip_runtime.h>
#include <hip/hip_bf16.h>

#define BGRAPH 32
#define NPTS   1024
#define FIN    6
#define FP     8      // padded feature dim (WMMA B operand, lanes >= FP masked to 0)
#define FOUT   128
#define KCH    6
#define NCLS   10

typedef __attribute__((ext_vector_type(2))) float v2f;
typedef __attribute__((ext_vector_type(8))) float v8f;

// ---------------------------------------------------------------------------
// Pad x [B*N,6] -> Tx0 [B*N,8] (features 6,7 zeroed)
// ---------------------------------------------------------------------------
__global__ void pad_tx0(const float* __restrict__ x, float* __restrict__ tx0) {
  int r = blockIdx.x * blockDim.x + threadIdx.x;
  if (r < BGRAPH * NPTS) {
#pragma unroll
    for (int f = 0; f < FP; ++f)
      tx0[(size_t)r * FP + f] = (f < FIN) ? x[(size_t)r * FIN + f] : 0.f;
  }
}

// ---------------------------------------------------------------------------
// One block per (graph, row). Computes A[i,:] = exp(-d2), zero diag, and
// dinv[i] = rsqrt(sum_j A[i,j]) (0 if deg==0). Coalesced row-major stores.
// ---------------------------------------------------------------------------
__global__ __launch_bounds__(256) void adj_rows(const float* __restrict__ x,
                                                float* __restrict__ A,
                                                float* __restrict__ dinv) {
  __shared__ float red[256];
  const int b = blockIdx.x / NPTS;
  const int i = blockIdx.x % NPTS;
  const float* xb = x + (size_t)b * NPTS * FIN;

  float xi[FIN];
  float sqi = 0.f;
#pragma unroll
  for (int f = 0; f < FIN; ++f) { xi[f] = xb[(size_t)i * FIN + f]; sqi += xi[f] * xi[f]; }

  float* Arow = A + ((size_t)b * NPTS + i) * NPTS;
  float dsum = 0.f;
#pragma unroll
  for (int s = 0; s < NPTS / 256; ++s) {
    const int j = threadIdx.x + s * 256;
    float dot = 0.f, sqj = 0.f;
#pragma unroll
    for (int f = 0; f < FIN; ++f) { const float xj = xb[(size_t)j * FIN + f]; dot += xi[f] * xj; sqj += xj * xj; }
    const float d2 = sqi - 2.f * dot + sqj;
    const float a = (j == i) ? 0.f : __expf(-d2);
    Arow[j] = a;
    dsum += a;
  }
  red[threadIdx.x] = dsum;
  __syncthreads();
  for (int off = 128; off > 0; off >>= 1) {
    if (threadIdx.x < off) red[threadIdx.x] += red[threadIdx.x + off];
    __syncthreads();
  }
  if (threadIdx.x == 0) {
    const float deg = red[0];
    dinv[(size_t)b * NPTS + i] = (deg > 0.f) ? rsqrtf(deg) : 0.f;
  }
}

// ---------------------------------------------------------------------------
// In-place rescale A -> L = -dinv_i * A_ij * dinv_j
// ---------------------------------------------------------------------------
__global__ void scale_L(float* __restrict__ A, const float* __restrict__ dinv) {
  const size_t idx = (size_t)blockIdx.x * blockDim.x + threadIdx.x;
  const size_t total = (size_t)BGRAPH * NPTS * NPTS;
  if (idx >= total) return;
  const size_t bn = idx / NPTS;          // b*NPTS + i
  const int j = (int)(idx % NPTS);
  const size_t b = bn / NPTS;
  const float di = dinv[bn];
  const float dj = dinv[b * NPTS + j];
  A[idx] = -di * dj * A[idx];
}

// ---------------------------------------------------------------------------
// One Chebyshev step: txNext = c2 * (L @ txPrev) - c0 * txPP   (c2,c0)=(1,0) k==1 else (2,1)
// and out (+)= txNext @ W_cheb[k]  (plus Tx0 @ W_cheb[0] on the first step).
// L-apply runs on V_WMMA_F32_16X16X4_F32: wave = one 16-row tile, K stepped by 4.
// Block = 8 waves sharing one graph; full [1024,8] txPrev staged in LDS (32 KB).
// A-fragment lane map: (m=lane%16, k=2*(lane/16)+vgpr)  -> 8B loads of L.
// B-fragment lane map: (k=2*(lane/16)+vgpr, n=lane%16)  -> LDS reads, n>=8 masked 0.
// D/C lane map: vgpr r, lane l -> (row = r + 8*(l/16), feat = l%16).
// ---------------------------------------------------------------------------
__global__ __launch_bounds__(256) void cheb_step(const float* __restrict__ L,
                                                 const float* __restrict__ txPrev,
                                                 const float* __restrict__ txPP,
                                                 float* __restrict__ txNext,
                                                 const float* __restrict__ Wcheb,
                                                 float* __restrict__ outw,
                                                 int k) {
  __shared__ float ldsTx[NPTS * FP];        // 32 KB: full-graph txPrev
  __shared__ float ldsTk[8][16 * FP];       // 4 KB: per-wave txNext tile

  const int tid  = threadIdx.x;
  const int wave = tid >> 5;
  const int lane = tid & 31;
  const int b    = blockIdx.x >> 3;                      // 8 blocks per graph
  const int tileInGraph = (blockIdx.x & 7) * 8 + wave;   // 0..63
  const int row0 = tileInGraph * 16;

  // Stage txPrev[b] into LDS (2048 float4, uniform trip count -> EXEC stays full)
  const float4* src = (const float4*)(txPrev + (size_t)b * NPTS * FP);
  float4* dst = (float4*)ldsTx;
#pragma unroll
  for (int t = tid; t < NPTS * FP / 4; t += 256) dst[t] = src[t];
  __syncthreads();

  const int   m   = lane & 15;
  const int   kh  = lane >> 4;                   // 0/1: K half
  const float sel = (m < FP) ? 1.f : 0.f;        // mask padded B columns, branch-free
  const int   nn  = m & (FP - 1);

  const float* Lrow = L + ((size_t)b * NPTS + row0 + m) * NPTS;

  v8f acc = {0.f, 0.f, 0.f, 0.f, 0.f, 0.f, 0.f, 0.f};
#pragma unroll 4
  for (int j = 0; j < NPTS; j += 4) {
    const int k0 = j + 2 * kh;
    v2f a;
    a.x = Lrow[k0];
    a.y = Lrow[k0 + 1];
    v2f bb;
    bb.x = sel * ldsTx[k0 * FP + nn];
    bb.y = sel * ldsTx[(k0 + 1) * FP + nn];
    acc = __builtin_amdgcn_wmma_f32_16x16x4_f32(false, a, false, bb,
                                                (short)0, acc, false, false);
  }

  // Chebyshev combine + store txNext (+ stage tile to LDS for projection)
  const int   first = (k == 1);
  const float c2 = first ? 1.f : 2.f;
  const float c0 = first ? 0.f : 1.f;
#pragma unroll
  for (int r = 0; r < 8; ++r) {
    const int row = row0 + r + 8 * kh;
    float v = c2 * acc[r];
    if (m < FP) {
      const size_t gi = ((size_t)b * NPTS + row) * FP + m;
      v -= c0 * txPP[gi];
      txNext[gi] = v;
      ldsTk[wave][(r + 8 * kh) * FP + m] = v;
    }
  }
  __syncthreads();

  // Projection: out(16x128) (+)= txk(16x6) @ Wk(6x128); first pass also adds Tx0 @ W0
  const float* Wk = Wcheb + (size_t)k * FIN * FOUT;
  const float* W0 = Wcheb;
#pragma unroll
  for (int cc = 0; cc < FOUT / 32; ++cc) {
    const int c = lane + 32 * cc;
    for (int rr = 0; rr < 16; ++rr) {
      float v = 0.f;
#pragma unroll
      for (int f = 0; f < FIN; ++f)
        v += ldsTk[wave][rr * FP + f] * Wk[f * FOUT + c];
      const size_t oi = ((size_t)b * NPTS + row0 + rr) * FOUT + c;
      if (first) {
#pragma unroll
        for (int f = 0; f < FIN; ++f)
          v += ldsTx[(row0 + rr) * FP + f] * W0[f * FOUT + c];  // ldsTx == Tx0 when k==1
        outw[oi] = v;
      } else {
        outw[oi] += v;
      }
    }
  }
}

// ---------------------------------------------------------------------------
// bias + ReLU + max-pool over N, then FC to 10 classes. One block per graph.
// ---------------------------------------------------------------------------
__global__ __launch_bounds__(128) void pool_fc(const float* __restrict__ outw,
                                               const float* __restrict__ bch,
                                               const float* __restrict__ Wfc,
                                               const float* __restrict__ bfc,
                                               float* __restrict__ y) {
  __shared__ float pooled[FOUT];
  const int b = blockIdx.x;
  const int f = threadIdx.x;
  const float bias = bch[f];
  float mx = 0.f;  // relu outputs are >= 0, so 0 is the identity for this max
  const float* p = outw + (size_t)b * NPTS * FOUT + f;
  for (int n = 0; n < NPTS; ++n) {
    float v = p[(size_t)n * FOUT] + bias;
    v = v > 0.f ? v : 0.f;
    mx = v > mx ? v : mx;
  }
  pooled[f] = mx;
  __syncthreads();
  if (f < NCLS) {
    float acc = bfc[f];
#pragma unroll 8
    for (int ff = 0; ff < FOUT; ++ff) acc += pooled[ff] * Wfc[ff * NCLS + f];
    y[b * NCLS + f] = acc;
  }
}

// ---------------------------------------------------------------------------
extern "C" void kernel_launch(void* const* d_in, const int* in_sizes, int n_in,
                              void* d_out, int out_size, void* d_ws, size_t ws_size,
                              hipStream_t stream) {
  const float* x     = (const float*)d_in[0];
  // d_in[1] = batch ids (int64) -- contiguous by construction, unused
  const float* Wcheb = (const float*)d_in[2];
  const float* bch   = (const float*)d_in[3];
  const float* Wfc   = (const float*)d_in[4];
  const float* bfc   = (const float*)d_in[5];
  float* y = (float*)d_out;

  // Workspace carve-up (fp32): L (134MB) | dinv | tx0..tx2 | out
  float* base = (float*)d_ws;
  size_t p = 0;
  float* Lw   = base + p; p += (size_t)BGRAPH * NPTS * NPTS;
  float* dinv = base + p; p += (size_t)BGRAPH * NPTS;
  float* tx0  = base + p; p += (size_t)BGRAPH * NPTS * FP;
  float* tx1  = base + p; p += (size_t)BGRAPH * NPTS * FP;
  float* tx2  = base + p; p += (size_t)BGRAPH * NPTS * FP;
  float* outw = base + p; p += (size_t)BGRAPH * NPTS * FOUT;

  pad_tx0<<<(BGRAPH * NPTS + 255) / 256, 256, 0, stream>>>(x, tx0);
  adj_rows<<<BGRAPH * NPTS, 256, 0, stream>>>(x, Lw, dinv);
  {
    const size_t total = (size_t)BGRAPH * NPTS * NPTS;
    scale_L<<<(unsigned)((total + 255) / 256), 256, 0, stream>>>(Lw, dinv);
  }

  float* bufs[3] = {tx0, tx1, tx2};
  int prev = 0, pp = 0;
  for (int k = 1; k < KCH; ++k) {
    const int next = (k == 1) ? 1 : (3 - prev - pp);
    cheb_step<<<BGRAPH * 8, 256, 0, stream>>>(Lw, bufs[prev], bufs[pp], bufs[next],
                                              Wcheb, outw, k);
    pp = prev; prev = next;
  }

  pool_fc<<<BGRAPH, 128, 0, stream>>>(outw, bch, Wfc, bfc, y);
}